// YoloLayer_42142219108678
// MI455X (gfx1250) — compile-verified
//
#include <hip/hip_runtime.h>

#ifndef __has_builtin
#define __has_builtin(x) 0
#endif

#define ATT  85     // 5 + 80 attributes per anchor
#define NA   3      // anchors
#define TILE 256    // spatial positions per block

typedef float v4f __attribute__((ext_vector_type(4)));

// ---------------------------------------------------------------------------
// gfx1250 async LDS->global path (ASYNCcnt).  Probe-learned signature for
// b128: (AS(1) int __vector(4)*, AS(3) int __vector(4)*, imm offset, imm cpol)
// ---------------------------------------------------------------------------
#if defined(__AMDGCN__) && __has_builtin(__builtin_amdgcn_global_store_async_from_lds_b128)
#define HAS_ASYNC_ST 1
typedef int v4i __attribute__((vector_size(16)));
typedef __attribute__((address_space(1))) v4i glb_v4i_t;
typedef __attribute__((address_space(3))) v4i lds_v4i_t;
#else
#define HAS_ASYNC_ST 0
#endif

__device__ __forceinline__ float fsigmoid(float v) {
    // v_exp_f32 + v_rcp_f32: ~2.5 ulp, plenty for a (0,1) sigmoid
    return __fdividef(1.0f, 1.0f + __expf(-v));
}

__global__ __launch_bounds__(TILE)
void yolo_head_kernel(const float* __restrict__ x,
                      const float* __restrict__ anchors,
                      const int*   __restrict__ img_h_p,
                      const int*   __restrict__ img_w_p,
                      float*       __restrict__ out)
{
    constexpr int H  = 76, W = 76;
    constexpr int HW = H * W;                       // 5776
    constexpr int TILES = (HW + TILE - 1) / TILE;   // 23

    const int tid  = threadIdx.x;
    const int tile = blockIdx.x % TILES;            // constant div -> mul/shift
    const int map  = blockIdx.x / TILES;            // b*NA + a
    const int a    = map % NA;
    const int s0   = tile * TILE;
    const int valid = min(TILE, HW - s0);

    const float stride_w = (float)(img_w_p[0] / W); // 8
    const float stride_h = (float)(img_h_p[0] / H); // 8
    // bw = exp(p)*(anc/stride)*stride = exp(p)*anc  (strides cancel)
    const float anc_w = anchors[2 * a + 0];
    const float anc_h = anchors[2 * a + 1];

    // Transposed tile: tilebuf[ds*ATT + c] == linear output order.  85 KB.
    __shared__ __align__(16) float tilebuf[TILE * ATT];

    // -------- Phase 1: coalesced load -> transform -> LDS transpose --------
    // Lanes consecutive in s => 128B-coalesced reads per channel.  Channel c
    // is the *loop index* => wave-uniform: special channels peeled, the hot
    // c=4..84 loop is branch-free sigmoid.  LDS write stride = 85 words
    // (odd) => conflict-free across 64 banks.
    const float* __restrict__ inBase = x + (size_t)map * ATT * HW + s0 + tid;
    if (tid < valid) {
        const int   s    = s0 + tid;
        const int   sy   = s / W;                   // constant div
        const float xoff = (float)(s - sy * W);
        const float yoff = (float)sy;
        float* __restrict__ row = &tilebuf[tid * ATT];

        row[0] = (fsigmoid(__builtin_nontemporal_load(inBase + 0 * HW)) + xoff) * stride_w;
        row[1] = (fsigmoid(__builtin_nontemporal_load(inBase + 1 * HW)) + yoff) * stride_h;
        row[2] = __expf(__builtin_nontemporal_load(inBase + 2 * HW)) * anc_w;
        row[3] = __expf(__builtin_nontemporal_load(inBase + 3 * HW)) * anc_h;
        #pragma unroll 9
        for (int c = 4; c < ATT; ++c)
            row[c] = fsigmoid(__builtin_nontemporal_load(inBase + (size_t)c * HW));
    }
    __syncthreads();

    // -------- Phase 2: pure linear LDS -> global streaming copy ------------
    // Contiguous output region; (valid*85) % 4 == 0 and outBase is 16B
    // aligned.  Async b128 stores: LDS -> memory DMA, no VGPR staging,
    // ASYNCcnt drained before kernel end.
    float* __restrict__ outBase = out + ((size_t)map * HW + (size_t)s0) * ATT;
    const int total = valid * ATT;
#if HAS_ASYNC_ST
    for (int k = tid * 4; k < total; k += TILE * 4) {
        __builtin_amdgcn_global_store_async_from_lds_b128(
            (glb_v4i_t*)(outBase + k),
            (lds_v4i_t*)(&tilebuf[k]),
            /*offset=*/0, /*cpol=*/0);
    }
  #if __has_builtin(__builtin_amdgcn_s_wait_asynccnt)
    __builtin_amdgcn_s_wait_asynccnt(0);
  #else
    asm volatile("s_wait_asynccnt 0x0" ::: "memory");
  #endif
#else
    for (int k = tid * 4; k < total; k += TILE * 4) {
        const v4f v = *(const v4f*)&tilebuf[k];
        __builtin_nontemporal_store(v, (v4f*)(outBase + k));
    }
#endif
}

extern "C" void kernel_launch(void* const* d_in, const int* in_sizes, int n_in,
                              void* d_out, int out_size, void* d_ws, size_t ws_size,
                              hipStream_t stream) {
    const float* x       = (const float*)d_in[0];
    const float* anchors = (const float*)d_in[1];
    const int*   img_h   = (const int*)d_in[2];
    const int*   img_w   = (const int*)d_in[3];
    float*       out     = (float*)d_out;

    const int H = 76, W = 76, HW = H * W;
    const int B = in_sizes[0] / (NA * ATT * HW);    // 16
    const int tiles = (HW + TILE - 1) / TILE;       // 23

    dim3 grid(B * NA * tiles);                      // 1104 blocks
    yolo_head_kernel<<<grid, TILE, 0, stream>>>(x, anchors, img_h, img_w, out);
}